// TUPEMultiheadAttention_20401094656390
// MI455X (gfx1250) — compile-verified
//
#include <hip/hip_runtime.h>
#include <math.h>

typedef float v2f __attribute__((ext_vector_type(2)));
typedef float v8f __attribute__((ext_vector_type(8)));

#define BATCH 4
#define SEQ   2048
#define INDIM 512
#define EMBED 512
#define NH    8
#define HD    64
#define MTOK  (BATCH * SEQ)
#define QB    64
#define KB    16

#define ATTN_SCALE 0.08838834764831845f  /* 1/sqrt(2*64) */

static __device__ __forceinline__ v8f wmma_f32(v2f a, v2f b, v8f c) {
  // V_WMMA_F32_16X16X4_F32 : D = A(16x4,f32) x B(4x16,f32) + C(16x16,f32)
  return __builtin_amdgcn_wmma_f32_16x16x4_f32(false, a, false, b, (short)0, c,
                                               false, false);
}

// ---------------------------------------------------------------------------
// C[M,N] = A[M,K] @ W[N,K]^T   (fp32 WMMA 16x16x4)
// grid: (M/32, N/256), block = 128 threads = 4 waves.
// Each wave: 32 rows x 64 cols = 8 accumulator tiles; per K-step 2 A-frags +
// 4 B-frags feed 8 WMMAs. K-loop is explicitly software-pipelined one step
// ahead so the 6 loads stay in flight behind the 8 WMMAs.
// A-frag (lane): A[row + (lane&15)][k + 2*(lane>>4) + {0,1}]
// B-frag (lane): W[col + (lane&15)][k + 2*(lane>>4) + {0,1}]
// C/D  (lane):   vgpr r holds C[row + r + 8*(lane>>4)][col + (lane&15)]
// ---------------------------------------------------------------------------
__global__ __launch_bounds__(128) void gemm_xwT(
    const float* __restrict__ A, const float* __restrict__ W,
    float* __restrict__ C, int M, int N, int K)
{
  const int lane = threadIdx.x & 31;
  const int wave = threadIdx.x >> 5;
  const int half = lane >> 4;
  const int l16  = lane & 15;
  const int row0 = blockIdx.x * 32;
  const int col0 = blockIdx.y * 256 + wave * 64;

  const float* ap0 = A + (size_t)(row0 + l16) * K + 2 * half;
  const float* ap1 = ap0 + (size_t)16 * K;
  const float* wp0 = W + (size_t)(col0 + l16) * K + 2 * half;
  const float* wp1 = wp0 + (size_t)16 * K;
  const float* wp2 = wp0 + (size_t)32 * K;
  const float* wp3 = wp0 + (size_t)48 * K;

  v8f acc00 = {}, acc01 = {}, acc02 = {}, acc03 = {};
  v8f acc10 = {}, acc11 = {}, acc12 = {}, acc13 = {};

  v2f a0 = *(const v2f*)(ap0);
  v2f a1 = *(const v2f*)(ap1);
  v2f b0 = *(const v2f*)(wp0);
  v2f b1 = *(const v2f*)(wp1);
  v2f b2 = *(const v2f*)(wp2);
  v2f b3 = *(const v2f*)(wp3);

  for (int k = 4; k < K; k += 4) {
    // prefetch next K-step fragments (kept independent of current WMMAs)
    v2f an0 = *(const v2f*)(ap0 + k);
    v2f an1 = *(const v2f*)(ap1 + k);
    v2f bn0 = *(const v2f*)(wp0 + k);
    v2f bn1 = *(const v2f*)(wp1 + k);
    v2f bn2 = *(const v2f*)(wp2 + k);
    v2f bn3 = *(const v2f*)(wp3 + k);

    acc00 = wmma_f32(a0, b0, acc00);
    acc01 = wmma_f32(a0, b1, acc01);
    acc02 = wmma_f32(a0, b2, acc02);
    acc03 = wmma_f32(a0, b3, acc03);
    acc10 = wmma_f32(a1, b0, acc10);
    acc11 = wmma_f32(a1, b1, acc11);
    acc12 = wmma_f32(a1, b2, acc12);
    acc13 = wmma_f32(a1, b3, acc13);

    a0 = an0; a1 = an1;
    b0 = bn0; b1 = bn1; b2 = bn2; b3 = bn3;
  }
  acc00 = wmma_f32(a0, b0, acc00);
  acc01 = wmma_f32(a0, b1, acc01);
  acc02 = wmma_f32(a0, b2, acc02);
  acc03 = wmma_f32(a0, b3, acc03);
  acc10 = wmma_f32(a1, b0, acc10);
  acc11 = wmma_f32(a1, b1, acc11);
  acc12 = wmma_f32(a1, b2, acc12);
  acc13 = wmma_f32(a1, b3, acc13);

#pragma unroll
  for (int r = 0; r < 8; ++r) {
    const int m = r + 8 * half;
    float* crow0 = C + (size_t)(row0 + m) * N + col0 + l16;
    crow0[0]  = acc00[r];
    crow0[16] = acc01[r];
    crow0[32] = acc02[r];
    crow0[48] = acc03[r];
    float* crow1 = crow0 + (size_t)16 * N;
    crow1[0]  = acc10[r];
    crow1[16] = acc11[r];
    crow1[32] = acc12[r];
    crow1[48] = acc13[r];
  }
}

// ---------------------------------------------------------------------------
// Flash attention with TUPE scores.
//   qkv : [B*S, 1536]  (head h: q = h*192+0..63, k = +64, v = +128)
//   pe  : [B*S, 1024]  (head h: uq = h*128+0..63, uk = +64)
//   per : [H, S, S]
//   vals: [B*S, 512]   (channel = h*64 + d)
// grid: (S/QB, B*H), block = 128 (4 waves); wave w owns queries [q0+16w, +16)
// ---------------------------------------------------------------------------
__global__ __launch_bounds__(128) void attn_kernel(
    const float* __restrict__ qkv, const float* __restrict__ pe,
    const float* __restrict__ per, float* __restrict__ vals)
{
  __shared__ float qs [QB][HD];     // queries, row-major
  __shared__ float uqs[QB][HD];     // pos-queries, row-major
  __shared__ float ks [KB][HD];     // keys, row-major
  __shared__ float uks[KB][HD];     // pos-keys, row-major
  __shared__ float vts[HD][KB];     // values, TRANSPOSED [d][key]
  __shared__ float wsc[4][16 * 16]; // per-wave score / P scratch
  __shared__ float wrow[4][16];     // per-wave per-row alpha / 1/l

  const int bh   = blockIdx.y;
  const int b    = bh / NH;
  const int h    = bh % NH;
  const int q0   = blockIdx.x * QB;
  const int tid  = threadIdx.x;
  const int lane = tid & 31;
  const int wave = tid >> 5;
  const int half = lane >> 4;
  const int l16  = lane & 15;

  const size_t tokbase = (size_t)b * SEQ;
  const int qoff = h * 192;        // q channels in qkv row
  const int uoff = h * 128;        // uq channels in pe row

  // stage the whole 64-query block (q and uq) once
  for (int i = tid; i < QB * HD; i += 128) {
    const int r = i >> 6, c = i & 63;
    qs [r][c] = qkv[(tokbase + q0 + r) * 1536 + qoff + c];
    uqs[r][c] = pe [(tokbase + q0 + r) * 1024 + uoff + c];
  }

  v8f acc[4] = {v8f{}, v8f{}, v8f{}, v8f{}};   // out tile: 16 q x 64 d
  float mrun = -1.0e30f;                        // valid in lanes 0..15
  float lrun = 0.0f;

  const int   qw  = q0 + wave * 16;
  const float* prw = per + ((size_t)h * SEQ + qw) * SEQ;
  float* sw = wsc[wave];

  for (int kb = 0; kb < SEQ; kb += KB) {
    __syncthreads();   // previous iteration's LDS reads done before overwrite
    for (int i = tid; i < KB * HD; i += 128) {
      const int r = i >> 6, c = i & 63;
      const size_t rowq = (tokbase + kb + r) * 1536 + qoff;
      ks [r][c] = qkv[rowq + 64 + c];
      vts[c][r] = qkv[rowq + 128 + c];
      uks[r][c] = pe [(tokbase + kb + r) * 1024 + uoff + 64 + c];
    }
    __syncthreads();

    // ---- scores: word q.k^T + pos uq.uk^T, 16x16 tile, 32 WMMAs ----
    v8f sc = {};
#pragma unroll
    for (int kk = 0; kk < HD; kk += 4) {
      v2f a  = *(const v2f*)&qs[wave * 16 + l16][kk + 2 * half];
      v2f bf = *(const v2f*)&ks[l16][kk + 2 * half];
      sc = wmma_f32(a, bf, sc);
    }
#pragma unroll
    for (int kk = 0; kk < HD; kk += 4) {
      v2f a  = *(const v2f*)&uqs[wave * 16 + l16][kk + 2 * half];
      v2f bf = *(const v2f*)&uks[l16][kk + 2 * half];
      sc = wmma_f32(a, bf, sc);
    }

    // ---- scale + PE_r bias, spill score tile to per-wave LDS scratch ----
    const float* prb = prw + kb;
#pragma unroll
    for (int r = 0; r < 8; ++r) {
      const int m = r + 8 * half;
      sw[m * 16 + l16] = sc[r] * ATTN_SCALE + prb[(size_t)m * SEQ + l16];
    }
    asm volatile("s_wait_dscnt 0" ::: "memory");

    // ---- online softmax: lanes 0..15 each own one query row ----
    if (lane < 16) {
      float v[16];
      float mt = -1.0e30f;
#pragma unroll
      for (int j = 0; j < 16; ++j) { v[j] = sw[lane * 16 + j]; mt = fmaxf(mt, v[j]); }
      const float mnew  = fmaxf(mrun, mt);
      const float alpha = __expf(mrun - mnew);
      float lsum = 0.0f;
#pragma unroll
      for (int j = 0; j < 16; ++j) {
        const float p = __expf(v[j] - mnew);
        sw[lane * 16 + j] = p;         // P tile replaces scores
        lsum += p;
      }
      lrun = lrun * alpha + lsum;
      mrun = mnew;
      wrow[wave][lane] = alpha;
    }
    asm volatile("s_wait_dscnt 0" ::: "memory");

    // ---- rescale accumulators by per-row alpha ----
    float fr[8];
#pragma unroll
    for (int r = 0; r < 8; ++r) fr[r] = wrow[wave][r + 8 * half];
#pragma unroll
    for (int t = 0; t < 4; ++t)
#pragma unroll
      for (int r = 0; r < 8; ++r) acc[t][r] *= fr[r];

    // ---- P @ V : 16 WMMAs (A = P from LDS, B = V^T tile from LDS) ----
#pragma unroll
    for (int kk = 0; kk < 4; ++kk) {
      v2f a = *(const v2f*)&sw[l16 * 16 + kk * 4 + 2 * half];
#pragma unroll
      for (int t = 0; t < 4; ++t) {
        v2f bf = *(const v2f*)&vts[t * 16 + l16][kk * 4 + 2 * half];
        acc[t] = wmma_f32(a, bf, acc[t]);
      }
    }
  }

  // ---- finalize: divide by row sums, write [b, s, h*64 + d] ----
  if (lane < 16) wrow[wave][lane] = 1.0f / lrun;
  asm volatile("s_wait_dscnt 0" ::: "memory");
  float fr[8];
#pragma unroll
  for (int r = 0; r < 8; ++r) fr[r] = wrow[wave][r + 8 * half];
#pragma unroll
  for (int t = 0; t < 4; ++t)
#pragma unroll
    for (int r = 0; r < 8; ++r) {
      const int m = r + 8 * half;
      vals[(tokbase + qw + m) * (size_t)EMBED + h * 64 + t * 16 + l16] =
          acc[t][r] * fr[r];
    }
}

// ---------------------------------------------------------------------------
extern "C" void kernel_launch(void* const* d_in, const int* in_sizes, int n_in,
                              void* d_out, int out_size, void* d_ws, size_t ws_size,
                              hipStream_t stream) {
  (void)in_sizes; (void)n_in; (void)out_size; (void)ws_size;
  const float* x    = (const float*)d_in[0];   // [B,S,512]
  const float* PE   = (const float*)d_in[1];   // [B,S,512]
  const float* PEr  = (const float*)d_in[2];   // [H,S,S]
  const float* Wqkv = (const float*)d_in[3];   // [1536,512]
  const float* UqUk = (const float*)d_in[4];   // [1024,512]
  const float* Wo   = (const float*)d_in[5];   // [512,512]
  float* out = (float*)d_out;                  // [B,S,512]

  float* qkv_buf = (float*)d_ws;                       // 8192 x 1536
  float* pe_buf  = qkv_buf + (size_t)MTOK * 1536;      // 8192 x 1024
  float* vals    = pe_buf  + (size_t)MTOK * 1024;      // 8192 x 512

  const dim3 blk(128);
  gemm_xwT<<<dim3(MTOK / 32, 1536 / 256), blk, 0, stream>>>(x, Wqkv, qkv_buf,
                                                            MTOK, 1536, INDIM);
  gemm_xwT<<<dim3(MTOK / 32, 1024 / 256), blk, 0, stream>>>(PE, UqUk, pe_buf,
                                                            MTOK, 1024, EMBED);
  attn_kernel<<<dim3(SEQ / QB, BATCH * NH), blk, 0, stream>>>(qkv_buf, pe_buf,
                                                              PEr, vals);
  gemm_xwT<<<dim3(MTOK / 32, EMBED / 256), blk, 0, stream>>>(vals, Wo, out,
                                                             MTOK, EMBED, EMBED);
}